// GeoGraphSAGE_44306882625629
// MI455X (gfx1250) — compile-verified
//
#include <hip/hip_runtime.h>
#include <hip/hip_bf16.h>

// GraphSAGE (3 layers, mean aggr) for MI455X / gfx1250.
// fp32 throughout (matches reference). Dense GEMMs via V_WMMA_F32_16X16X4_F32.
// Aggregation: COO atomic scatter-add, L2-resident (working set ~100MB < 192MB L2).

#define NN   100000
#define EE   1600000
#define INC  32
#define HC   64
#define OUTC 16

typedef float v2f __attribute__((ext_vector_type(2)));
typedef float v8f __attribute__((ext_vector_type(8)));

// ---------------------------------------------------------------------------
// WMMA GEMM: out[nrows x NT*16] = (SCALE1? diag(rdeg):I) * A1 @ B1
//                                 (+ A2 @ B2 if DUAL) + bias, optional ReLU.
// One wave computes a 16-row x (NT*16)-col tile. K is the A row stride.
// In-place out==A1 is safe: a wave stores only after all its A reads finish,
// and row blocks are disjoint across waves.
// ---------------------------------------------------------------------------
template<int K, int NT, bool DUAL, bool SCALE1, bool RELU>
__global__ __launch_bounds__(128) void gemm_wmma_k(
    const float* __restrict__ A1, const float* __restrict__ B1,
    const float* __restrict__ A2, const float* __restrict__ B2,
    const float* __restrict__ rdeg, const float* __restrict__ bias,
    float* __restrict__ out, int nrows)
{
    const int wid  = (int)blockIdx.x * (blockDim.x >> 5) + ((int)threadIdx.x >> 5);
    const int lane = threadIdx.x & 31;
    const int row0 = wid << 4;
    if (row0 >= nrows) return;                 // uniform per wave: EXEC stays all-1s
    const int half = lane >> 4;                // 0: lanes 0-15, 1: lanes 16-31
    const int l16  = lane & 15;
    const int m    = row0 + l16;               // A-matrix row owned by this lane
    constexpr int LDB = NT * 16;

    v8f acc[NT];
#pragma unroll
    for (int t = 0; t < NT; ++t) acc[t] = (v8f){0,0,0,0,0,0,0,0};

    const float sc = SCALE1 ? rdeg[m] : 1.0f;  // fold mean-divide into A load
    const float* a1 = A1 + (size_t)m * K;
    const float* a2 = DUAL ? (A2 + (size_t)m * K) : nullptr;

#pragma unroll
    for (int k = 0; k < K; k += 4) {
        // A 16x4 f32 layout: lanes 0-15 hold K=k,k+1; lanes 16-31 hold K=k+2,k+3
        const int kb = k + half * 2;
        v2f af1; af1.x = a1[kb] * sc; af1.y = a1[kb + 1] * sc;
        v2f af2;
        if (DUAL) { af2.x = a2[kb]; af2.y = a2[kb + 1]; }
#pragma unroll
        for (int t = 0; t < NT; ++t) {
            v2f bf1;
            bf1.x = B1[(size_t)kb       * LDB + t * 16 + l16];
            bf1.y = B1[(size_t)(kb + 1) * LDB + t * 16 + l16];
            acc[t] = __builtin_amdgcn_wmma_f32_16x16x4_f32(
                false, af1, false, bf1, (short)0, acc[t], false, false);
            if (DUAL) {
                v2f bf2;
                bf2.x = B2[(size_t)kb       * LDB + t * 16 + l16];
                bf2.y = B2[(size_t)(kb + 1) * LDB + t * 16 + l16];
                acc[t] = __builtin_amdgcn_wmma_f32_16x16x4_f32(
                    false, af2, false, bf2, (short)0, acc[t], false, false);
            }
        }
    }

    // C/D layout: VGPR r -> row r (lanes 0-15) / row r+8 (lanes 16-31)
#pragma unroll
    for (int t = 0; t < NT; ++t) {
        const float bc = bias[t * 16 + l16];
#pragma unroll
        for (int r = 0; r < 8; ++r) {
            const int rr = r + half * 8;
            float v = acc[t][r] + bc;
            if (RELU) v = fmaxf(v, 0.0f);
            out[(size_t)(row0 + rr) * LDB + t * 16 + l16] = v;
        }
    }
}

// ---------------------------------------------------------------------------
// Degree (in-degree per dst), then reciprocal of max(deg,1)
// ---------------------------------------------------------------------------
__global__ void degree_k(const int* __restrict__ dst, float* __restrict__ deg, int e)
{
    int tid = blockIdx.x * blockDim.x + threadIdx.x;
    if (tid < e) atomicAdd(&deg[dst[tid]], 1.0f);
}

__global__ void rdeg_k(float* __restrict__ deg, int n)
{
    int tid = blockIdx.x * blockDim.x + threadIdx.x;
    if (tid < n) deg[tid] = 1.0f / fmaxf(deg[tid], 1.0f);
}

// ---------------------------------------------------------------------------
// Scatter-add: 16 threads per edge, float4 per thread (coalesced B128 gather,
// 4 f32 atomics). agg[dst] += h[src]. Gathers hit L2 (25.6MB node array).
// ---------------------------------------------------------------------------
__global__ __launch_bounds__(256) void scatter_k(
    const float* __restrict__ h, const int* __restrict__ src,
    const int* __restrict__ dst, float* __restrict__ agg)
{
    const unsigned tid = blockIdx.x * 256u + threadIdx.x;   // EE*16 threads exactly
    const unsigned e   = tid >> 4;
    const unsigned c4  = (tid & 15u) << 2;
    const int s = src[e];
    const int d = dst[e];
    const float4 v = *(const float4*)(h + (size_t)s * HC + c4);
    float* ap = agg + (size_t)d * HC + c4;
    atomicAdd(ap + 0, v.x);
    atomicAdd(ap + 1, v.y);
    atomicAdd(ap + 2, v.z);
    atomicAdd(ap + 3, v.w);
}

// ---------------------------------------------------------------------------
// Per-row LayerNorm(64) + ReLU + residual. One wave32 per row, 2 elems/lane.
// ---------------------------------------------------------------------------
__global__ __launch_bounds__(256) void ln_relu_res_k(
    const float* __restrict__ t, const float* __restrict__ g,
    const float* __restrict__ b, const float* __restrict__ xin,
    float* __restrict__ out, int nrows)
{
    const int wid  = (int)blockIdx.x * (blockDim.x >> 5) + ((int)threadIdx.x >> 5);
    const int lane = threadIdx.x & 31;
    if (wid >= nrows) return;
    const float* row = t + (size_t)wid * HC;
    float v0 = row[lane], v1 = row[lane + 32];
    float s = v0 + v1;
#pragma unroll
    for (int off = 16; off > 0; off >>= 1) s += __shfl_xor(s, off, 32);
    const float mu = s * (1.0f / HC);
    const float d0 = v0 - mu, d1 = v1 - mu;
    float q = d0 * d0 + d1 * d1;
#pragma unroll
    for (int off = 16; off > 0; off >>= 1) q += __shfl_xor(q, off, 32);
    const float rstd = rsqrtf(q * (1.0f / HC) + 1e-5f);
    const float y0 = fmaxf(d0 * rstd * g[lane]      + b[lane],      0.0f);
    const float y1 = fmaxf(d1 * rstd * g[lane + 32] + b[lane + 32], 0.0f);
    float* o = out + (size_t)wid * HC;
    const float* xi = xin + (size_t)wid * HC;
    o[lane]      = y0 + xi[lane];
    o[lane + 32] = y1 + xi[lane + 32];
}

// ---------------------------------------------------------------------------
extern "C" void kernel_launch(void* const* d_in, const int* in_sizes, int n_in,
                              void* d_out, int out_size, void* d_ws, size_t ws_size,
                              hipStream_t stream)
{
    const float* x      = (const float*)d_in[0];
    const int*   ei     = (const int*)d_in[1];      // [2,E]: src row then dst row
    const float* proj_w = (const float*)d_in[2];
    const float* proj_b = (const float*)d_in[3];
    const float* wl0 = (const float*)d_in[4];  const float* bl0 = (const float*)d_in[5];
    const float* wr0 = (const float*)d_in[6];  const float* g0  = (const float*)d_in[7];
    const float* be0 = (const float*)d_in[8];
    const float* wl1 = (const float*)d_in[9];  const float* bl1 = (const float*)d_in[10];
    const float* wr1 = (const float*)d_in[11]; const float* g1  = (const float*)d_in[12];
    const float* be1 = (const float*)d_in[13];
    const float* wl2 = (const float*)d_in[14]; const float* bl2 = (const float*)d_in[15];
    const float* wr2 = (const float*)d_in[16];

    const int* srcI = ei;
    const int* dstI = ei + EE;

    // Workspace layout (fp32): hA | hB | agg | deg  -> ~77.2 MB
    const size_t N64 = (size_t)NN * HC;
    float* hA  = (float*)d_ws;
    float* hB  = hA + N64;
    float* agg = hB + N64;
    float* deg = agg + N64;

    const int GEMM_BLK   = 128;                          // 4 waves/block
    const int gemm_grid  = (NN / 16 + 3) / 4;            // 6250 wave-tiles
    const int scat_grid  = (EE * 16) / 256;              // exact: 100000 blocks
    const int ln_grid    = NN / 8;                       // 8 waves(rows)/block
    const int deg_grid   = (EE + 255) / 256;
    const int rdeg_grid  = (NN + 255) / 256;

    // Degree (shared across all three aggregations), then reciprocal in place.
    hipMemsetAsync(deg, 0, NN * sizeof(float), stream);
    degree_k<<<deg_grid, 256, 0, stream>>>(dstI, deg, EE);
    rdeg_k<<<rdeg_grid, 256, 0, stream>>>(deg, NN);

    // h0 = relu(x @ proj_w + proj_b)   [K=32, 64 out]
    gemm_wmma_k<INC, 4, false, false, true><<<gemm_grid, GEMM_BLK, 0, stream>>>(
        x, proj_w, nullptr, nullptr, nullptr, proj_b, hA, NN);

    // ---- Layer 0: hA -> hB ----
    hipMemsetAsync(agg, 0, N64 * sizeof(float), stream);
    scatter_k<<<scat_grid, 256, 0, stream>>>(hA, srcI, dstI, agg);
    gemm_wmma_k<HC, 4, true, true, false><<<gemm_grid, GEMM_BLK, 0, stream>>>(
        agg, wl0, hA, wr0, deg, bl0, agg, NN);           // in-place: agg <- SAGE pre-norm
    ln_relu_res_k<<<ln_grid, 256, 0, stream>>>(agg, g0, be0, hA, hB, NN);

    // ---- Layer 1: hB -> hA ----
    hipMemsetAsync(agg, 0, N64 * sizeof(float), stream);
    scatter_k<<<scat_grid, 256, 0, stream>>>(hB, srcI, dstI, agg);
    gemm_wmma_k<HC, 4, true, true, false><<<gemm_grid, GEMM_BLK, 0, stream>>>(
        agg, wl1, hB, wr1, deg, bl1, agg, NN);
    ln_relu_res_k<<<ln_grid, 256, 0, stream>>>(agg, g1, be1, hB, hA, NN);

    // ---- Final layer: hA -> d_out (N x 16) ----
    hipMemsetAsync(agg, 0, N64 * sizeof(float), stream);
    scatter_k<<<scat_grid, 256, 0, stream>>>(hA, srcI, dstI, agg);
    gemm_wmma_k<HC, 1, true, true, false><<<gemm_grid, GEMM_BLK, 0, stream>>>(
        agg, wl2, hA, wr2, deg, bl2, (float*)d_out, NN);
}